// GroupAttention_83348135346330
// MI455X (gfx1250) — compile-verified
//
#include <hip/hip_runtime.h>

typedef __attribute__((ext_vector_type(2))) float v2f;
typedef __attribute__((ext_vector_type(8))) float v8f;

#define NUM_B   64
#define NUM_H   8
#define NUM_G   3
#define KPTOK   6
#define NTOK    262           // 256 patches + 6 keypoints
#define NKEFF   274           // 3*6 shared keypoints + 256 patches
#define NKPAD   288           // padded to 18 tiles of 16
#define DMODEL  512
#define DHEAD   64
#define MROWS   (NUM_B*NUM_G*NTOK)   // 50304 (exact multiple of 16)
#define QKVC    (3*DMODEL)           // 1536
#define KSTRIDE 68            // LDS row stride for k/v tiles (floats), bank-conflict free
#define ASTRIDE 292           // LDS row stride for attn rows (floats)
#define NEG_MAX (-3.402823466e38f)

__device__ __forceinline__ v8f wmma4(v2f a, v2f b, v8f c) {
  // V_WMMA_F32_16X16X4_F32 : D = A(16x4,f32) * B(4x16,f32) + C(16x16,f32)
  return __builtin_amdgcn_wmma_f32_16x16x4_f32(false, a, false, b, (short)0, c,
                                               false, false);
}

// Async 16-byte global->LDS copy (GLOBAL_LOAD_ASYNC_TO_LDS_B128, ASYNCcnt).
// LDS operand is the 32-bit LDS byte offset = low 32 bits of the generic addr.
__device__ __forceinline__ void async_copy_b128(const float* gsrc, float* ldst) {
  unsigned loff = (unsigned)(uintptr_t)ldst;
  asm volatile("global_load_async_to_lds_b128 %0, %1, off"
               :: "v"(loff), "v"(gsrc) : "memory");
}

__device__ __forceinline__ void wait_asynccnt0() {
  asm volatile("s_wait_asynccnt 0" ::: "memory");
}

// ---------------------------------------------------------------------------
// Generic fp32 WMMA GEMM: C[M x NCOLS] = A[M x KDIM] @ B[KDIM x NCOLS] (+bias)
// One wave computes a 16x64 tile of C (4 accumulators, A-fragment reused x4).
// ---------------------------------------------------------------------------
template<int KDIM, int NCOLS, bool HAS_BIAS>
__global__ __launch_bounds__(32)
void gemm_wmma(const float* __restrict__ A, const float* __restrict__ B,
               const float* __restrict__ bias, float* __restrict__ C) {
  const int m0   = blockIdx.x << 4;
  const int n0   = blockIdx.y << 6;
  const int lane = threadIdx.x;
  const int lm   = lane & 15;         // column / row-in-tile index
  const int kh   = (lane >> 4) << 1;  // K offset for upper half-wave (0 or 2)
  const int rh   = (lane >> 4) << 3;  // row offset for C layout (0 or 8)

  const float* arow = A + (long)(m0 + lm) * KDIM;
  v8f acc0 = {}, acc1 = {}, acc2 = {}, acc3 = {};

  for (int k = 0; k < KDIM; k += 4) {
    __builtin_prefetch(arow + k + 64, 0, 0);     // global_prefetch_b8 on A panel
    v2f af; af.x = arow[k + kh]; af.y = arow[k + kh + 1];
    const float* bp = B + (long)(k + kh) * NCOLS + n0 + lm;
    v2f b0; b0.x = bp[0];  b0.y = bp[NCOLS];
    v2f b1; b1.x = bp[16]; b1.y = bp[NCOLS + 16];
    v2f b2; b2.x = bp[32]; b2.y = bp[NCOLS + 32];
    v2f b3; b3.x = bp[48]; b3.y = bp[NCOLS + 48];
    acc0 = wmma4(af, b0, acc0);
    acc1 = wmma4(af, b1, acc1);
    acc2 = wmma4(af, b2, acc2);
    acc3 = wmma4(af, b3, acc3);
  }

  v8f accs[4] = {acc0, acc1, acc2, acc3};
#pragma unroll
  for (int t = 0; t < 4; ++t) {
    const int col = n0 + t * 16 + lm;
    const float bv = HAS_BIAS ? bias[col] : 0.0f;
#pragma unroll
    for (int e = 0; e < 8; ++e) {
      C[(long)(m0 + e + rh) * NCOLS + col] = accs[t][e] + bv;
    }
  }
}

// ---------------------------------------------------------------------------
// Fused attention per (b,h,g): async-stage gathered K/V into LDS once, then
// each wave processes 16-row query tiles: dots (WMMA) -> mask -> write dots ->
// softmax (LDS) -> attn@V (WMMA) -> attn_out.
// ---------------------------------------------------------------------------
__global__ __launch_bounds__(128)
void attention(const float* __restrict__ qkv,   // [B,G,N, 1536] (q|k|v)
               float* __restrict__ attn_out,    // [B,G,N, 512]
               float* __restrict__ dots_out) {  // [B,H,G, 262, 274]
  __shared__ float lds_k[NKPAD * KSTRIDE];
  __shared__ float lds_v[NKPAD * KSTRIDE];
  __shared__ float lds_attn[4][16 * ASTRIDE];

  const int bid = blockIdx.x;
  const int b   = bid / (NUM_H * NUM_G);
  const int hg  = bid % (NUM_H * NUM_G);
  const int h   = hg / NUM_G;
  const int g   = hg % NUM_G;
  const int tid = threadIdx.x;

  // ---- zero-fill K padding rows (274..287) while async loads stream in ----
  for (int idx = tid; idx < (NKPAD - NKEFF) * 16; idx += 128) {
    const int row = NKEFF + (idx >> 4);
    const int c4  = (idx & 15) << 2;
    const float4 z = make_float4(0.f, 0.f, 0.f, 0.f);
    *(float4*)(lds_k + row * KSTRIDE + c4) = z;
    *(float4*)(lds_v + row * KSTRIDE + c4) = z;
  }

  // ---- async-stage k_eff / v_eff (gathered keypoints + own patches) ----
  for (int idx = tid; idx < NKEFF * 16; idx += 128) {
    const int row = idx >> 4;
    const int c4  = (idx & 15) << 2;             // 0..60, float4 chunk
    int gg, tok;
    if (row < NUM_G * KPTOK) { gg = row / KPTOK; tok = row % KPTOK; }
    else                     { gg = g; tok = row - (NUM_G * KPTOK - KPTOK); }
    const float* src = qkv + ((long)(b * NUM_G + gg) * NTOK + tok) * QKVC
                     + h * DHEAD + c4;
    async_copy_b128(src + DMODEL,     lds_k + row * KSTRIDE + c4);  // K block
    async_copy_b128(src + 2 * DMODEL, lds_v + row * KSTRIDE + c4);  // V block
  }
  wait_asynccnt0();
  __syncthreads();

  const int wave = tid >> 5;
  const int lane = tid & 31;
  const int lm   = lane & 15;
  const int kh   = (lane >> 4) << 1;
  const int rh   = (lane >> 4) << 3;
  float* my_attn = lds_attn[wave];
  const float scale = 0.044194173824159216f;     // 512^-0.5

  for (int it = wave; it < 17; it += 4) {
    const int i0 = it << 4;
    const bool rows_full = (i0 + 16 <= NTOK);    // wave-uniform

    // ---- cache Q fragments for all 16 K-steps (Dh=64) ----
    v2f qf[16];
    {
      int rowi = i0 + lm; if (rowi > NTOK - 1) rowi = NTOK - 1;
      const float* qrow = qkv + ((long)(b * NUM_G + g) * NTOK + rowi) * QKVC
                        + h * DHEAD;
#pragma unroll
      for (int ks = 0; ks < 16; ++ks) {
        qf[ks].x = qrow[(ks << 2) + kh];
        qf[ks].y = qrow[(ks << 2) + kh + 1];
      }
    }

    // ---- dots = scale * q @ k_eff^T, mask, emit + keep in LDS ----
    for (int jt = 0; jt < 18; ++jt) {
      const int j0 = jt << 4;
      v8f acc = {};
#pragma unroll
      for (int ks = 0; ks < 16; ++ks) {
        const float* kp = lds_k + (j0 + lm) * KSTRIDE + (ks << 2) + kh;
        v2f bf; bf.x = kp[0]; bf.y = kp[1];
        acc = wmma4(qf[ks], bf, acc);
      }

      const int j = j0 + lm;
      float vals[8];
#pragma unroll
      for (int e = 0; e < 8; ++e) {
        const int i = i0 + e + rh;
        float val = acc[e] * scale;
        if (j == 0 || (g == 0 && i == 0) || j >= NKEFF) val = NEG_MAX;
        my_attn[(e + rh) * ASTRIDE + j] = val;
        vals[e] = val;
      }

      float* drow = dots_out
          + (((long)(b * NUM_H + h) * NUM_G + g) * NTOK + i0 + rh) * NKEFF + j;
      if (rows_full && (j0 + 16 <= NKEFF)) {
        // fast path: whole tile in-bounds, no per-lane EXEC juggling
#pragma unroll
        for (int e = 0; e < 8; ++e) drow[(long)e * NKEFF] = vals[e];
      } else {
#pragma unroll
        for (int e = 0; e < 8; ++e) {
          if ((i0 + e + rh < NTOK) && (j < NKEFF)) drow[(long)e * NKEFF] = vals[e];
        }
      }
    }

    // ---- row softmax (lane r handles row r; all-masked row -> uniform) ----
    if (lane < 16) {
      float* rowp = my_attn + lane * ASTRIDE;
      float m = NEG_MAX;
      for (int j = 0; j < NKEFF; ++j) m = fmaxf(m, rowp[j]);
      float s = 0.f;
      for (int j = 0; j < NKEFF; ++j) {
        const float e = __expf(rowp[j] - m);
        rowp[j] = e; s += e;
      }
      const float inv = 1.0f / s;
      for (int j = 0; j < NKEFF; ++j) rowp[j] *= inv;
      for (int j = NKEFF; j < ASTRIDE; ++j) rowp[j] = 0.f;  // zero K padding
    }

    // ---- out_tile[16x64] = attn[16x288] @ v_eff[288x64] ----
    v8f oacc[4] = {};
    for (int ks = 0; ks < NKPAD / 4; ++ks) {
      const float* ap = my_attn + lm * ASTRIDE + (ks << 2) + kh;
      v2f af; af.x = ap[0]; af.y = ap[1];
#pragma unroll
      for (int dt = 0; dt < 4; ++dt) {
        const float* vp = lds_v + ((ks << 2) + kh) * KSTRIDE + dt * 16 + lm;
        v2f bf; bf.x = vp[0]; bf.y = vp[KSTRIDE];
        oacc[dt] = wmma4(af, bf, oacc[dt]);
      }
    }

    float* orow = attn_out + ((long)(b * NUM_G + g) * NTOK + i0 + rh) * DMODEL
                + h * DHEAD + lm;
    if (rows_full) {
#pragma unroll
      for (int dt = 0; dt < 4; ++dt)
#pragma unroll
        for (int e = 0; e < 8; ++e)
          orow[(long)e * DMODEL + dt * 16] = oacc[dt][e];
    } else {
#pragma unroll
      for (int dt = 0; dt < 4; ++dt)
#pragma unroll
        for (int e = 0; e < 8; ++e)
          if (i0 + e + rh < NTOK) orow[(long)e * DMODEL + dt * 16] = oacc[dt][e];
    }
  }
}

// ---------------------------------------------------------------------------
extern "C" void kernel_launch(void* const* d_in, const int* in_sizes, int n_in,
                              void* d_out, int out_size, void* d_ws, size_t ws_size,
                              hipStream_t stream) {
  const float* x     = (const float*)d_in[0];  // [64,3,262,512]
  const float* w_qkv = (const float*)d_in[1];  // [512,1536]
  const float* w_out = (const float*)d_in[2];  // [512,512]
  const float* b_out = (const float*)d_in[3];  // [512]

  float* out  = (float*)d_out;                       // [64,3,262,512]
  float* dots = out + (long)MROWS * DMODEL;          // [64,8,3,262,274]

  float* qkv      = (float*)d_ws;                    // [MROWS, 1536]
  float* attn_out = qkv + (long)MROWS * QKVC;        // [MROWS, 512]

  // 1) QKV projection
  gemm_wmma<DMODEL, QKVC, false>
      <<<dim3(MROWS / 16, QKVC / 64), 32, 0, stream>>>(x, w_qkv, nullptr, qkv);

  // 2) Fused masked attention (also emits dots output)
  attention<<<dim3(NUM_B * NUM_H * NUM_G), 128, 0, stream>>>(qkv, attn_out, dots);

  // 3) Output projection + bias
  gemm_wmma<DMODEL, DMODEL, true>
      <<<dim3(MROWS / 16, DMODEL / 64), 32, 0, stream>>>(attn_out, w_out, b_out, out);
}